// PyGGCNModel_67216238182417
// MI455X (gfx1250) — compile-verified
//
#include <hip/hip_runtime.h>
#include <hip/hip_bf16.h>

#define NF 128   // input feature dim
#define NH 64    // hidden dim

typedef float v2f __attribute__((ext_vector_type(2)));
typedef float v8f __attribute__((ext_vector_type(8)));

__device__ __forceinline__ void atomic_add_f32(float* p, float v) {
    __hip_atomic_fetch_add(p, v, __ATOMIC_RELAXED, __HIP_MEMORY_SCOPE_AGENT);
}

// ---------------- degree / normalization ----------------

// deg[i] = 1.0f  (self-loop weight)
__global__ void k_deg_init(float* __restrict__ deg, int n) {
    int i = blockIdx.x * blockDim.x + threadIdx.x;
    if (i < n) deg[i] = 1.0f;
}

// deg[dst[e]] += ew[e]
__global__ void k_deg_accum(const int* __restrict__ dst, const float* __restrict__ ew,
                            float* __restrict__ deg, int E) {
    int e = blockIdx.x * blockDim.x + threadIdx.x;
    if (e < E) atomic_add_f32(&deg[dst[e]], ew[e]);
}

// dinv = deg > 0 ? rsqrt(max(deg,1e-12)) : 0   (in place)
__global__ void k_dinv(float* __restrict__ deg, int n) {
    int i = blockIdx.x * blockDim.x + threadIdx.x;
    if (i < n) {
        float d = deg[i];
        deg[i] = (d > 0.0f) ? rsqrtf(fmaxf(d, 1e-12f)) : 0.0f;
    }
}

// ---------------- layer 1 GEMM: xw = x @ W1 (WMMA f32 16x16x4) ----------------
// One block = 128 threads = 4 waves. Block handles a 16-row panel of x.
// Wave w computes the 16x16 output tile at columns [16w, 16w+16).
// A panel (16x128, 8KB) and all of W1 (128x64, 32KB) staged in LDS.
__global__ void __launch_bounds__(128) k_gemm1_wmma(const float* __restrict__ x,
                                                    const float* __restrict__ W1,
                                                    float* __restrict__ xw,
                                                    int n_tiles /* rows/16 */) {
    __shared__ float sA[16 * NF];   // 8 KB
    __shared__ float sW[NF * NH];   // 32 KB
    const int t  = threadIdx.x;
    const int m0 = blockIdx.x * 16;

    // stage A panel: 2048 floats = 512 float4
    {
        const float4* s4 = (const float4*)(x + (size_t)m0 * NF);
        float4* d4 = (float4*)sA;
        #pragma unroll
        for (int i = 0; i < 4; ++i) d4[i * 128 + t] = s4[i * 128 + t];
    }
    // stage W1: 8192 floats = 2048 float4
    {
        const float4* s4 = (const float4*)W1;
        float4* d4 = (float4*)sW;
        #pragma unroll
        for (int i = 0; i < 16; ++i) d4[i * 128 + t] = s4[i * 128 + t];
    }
    __syncthreads();

    const int wave = t >> 5;        // tile_n : 0..3
    const int lane = t & 31;
    const int half = lane >> 4;     // 0: K pair {0,1}; 1: K pair {2,3}
    const int l16  = lane & 15;     // row (A) / col (B,C/D)
    const int n0   = wave * 16;

    v8f acc = {0.f, 0.f, 0.f, 0.f, 0.f, 0.f, 0.f, 0.f};
    #pragma unroll
    for (int k = 0; k < NF; k += 4) {
        const int ka = k + half * 2;
        v2f a, b;
        // A 16x4 f32: VGPR0 = K={0|2}, VGPR1 = K={1|3}, lanes 0-15 rows, 16-31 rows again
        a.x = sA[l16 * NF + ka];
        a.y = sA[l16 * NF + ka + 1];
        // B 4x16 f32: VGPR0 = row K={0|2} across lanes, VGPR1 = row K={1|3}
        b.x = sW[ka * NH + n0 + l16];
        b.y = sW[(ka + 1) * NH + n0 + l16];
        acc = __builtin_amdgcn_wmma_f32_16x16x4_f32(
            false, a, false, b, (short)0, acc, false, false);
    }

    // C/D 16x16 f32: VGPR r -> M=r (lanes 0-15) / M=r+8 (lanes 16-31), N=l16
    float* o = xw + (size_t)(m0 + half * 8) * NH + n0 + l16;
    #pragma unroll
    for (int r = 0; r < 8; ++r) o[(size_t)r * NH] = acc[r];
}

// scalar tail for rows not covered by 16-row tiles (unused when N % 16 == 0)
__global__ void k_gemm1_tail(const float* __restrict__ x, const float* __restrict__ W1,
                             float* __restrict__ xw, int row0, int n_rows) {
    int i = blockIdx.x * blockDim.x + threadIdx.x;   // (row-row0)*NH + col
    int rows = n_rows - row0;
    if (i >= rows * NH) return;
    int r = row0 + i / NH, c = i % NH;
    float s = 0.f;
    for (int k = 0; k < NF; ++k) s += x[(size_t)r * NF + k] * W1[(size_t)k * NH + c];
    xw[(size_t)r * NH + c] = s;
}

// ---------------- layer 1 aggregation ----------------

// h1[n][f] = b1[f] + dinv[n]^2 * xw[n][f]   (bias + self-loop term)
__global__ void k_h1_init(const float* __restrict__ xw, const float* __restrict__ dinv,
                          const float* __restrict__ b1, float* __restrict__ h1, int n) {
    int i = blockIdx.x * blockDim.x + threadIdx.x;
    if (i >= n * NH) return;
    int node = i / NH, f = i % NH;
    float di = dinv[node];
    h1[i] = b1[f] + di * di * xw[i];
}

// one wave per edge; each lane handles 2 features
__global__ void k_scatter1(const int* __restrict__ src, const int* __restrict__ dst,
                           const float* __restrict__ ew, const float* __restrict__ dinv,
                           const float* __restrict__ xw, float* __restrict__ h1, int E) {
    int e = (blockIdx.x * blockDim.x + threadIdx.x) >> 5;
    if (e >= E) return;
    int lane = threadIdx.x & 31;
    int s = src[e], d = dst[e];
    float norm = dinv[s] * ew[e] * dinv[d];
    float2 v = *(const float2*)(xw + (size_t)s * NH + lane * 2);
    float* o = h1 + (size_t)d * NH + lane * 2;
    atomic_add_f32(o,     norm * v.x);
    atomic_add_f32(o + 1, norm * v.y);
}

// ---------------- layer 2 ----------------

// z[n] = sum_f relu(h1[n][f]) * W2[f]
__global__ void k_relu_dot(const float* __restrict__ h1, const float* __restrict__ W2,
                           float* __restrict__ z, int n) {
    int i = blockIdx.x * blockDim.x + threadIdx.x;
    if (i >= n) return;
    const float4* hp = (const float4*)(h1 + (size_t)i * NH);
    const float4* wp = (const float4*)W2;
    float s = 0.f;
    #pragma unroll
    for (int j = 0; j < NH / 4; ++j) {
        float4 h = hp[j], w = wp[j];
        s += fmaxf(h.x, 0.f) * w.x + fmaxf(h.y, 0.f) * w.y +
             fmaxf(h.z, 0.f) * w.z + fmaxf(h.w, 0.f) * w.w;
    }
    z[i] = s;
}

// out[n] = b2 + dinv[n]^2 * z[n]
__global__ void k_out_init(const float* __restrict__ z, const float* __restrict__ dinv,
                           const float* __restrict__ b2, float* __restrict__ out, int n) {
    int i = blockIdx.x * blockDim.x + threadIdx.x;
    if (i >= n) return;
    float di = dinv[i];
    out[i] = b2[0] + di * di * z[i];
}

// out[dst[e]] += dinv[src]*ew*dinv[dst] * z[src]
__global__ void k_scatter2(const int* __restrict__ src, const int* __restrict__ dst,
                           const float* __restrict__ ew, const float* __restrict__ dinv,
                           const float* __restrict__ z, float* __restrict__ out, int E) {
    int e = blockIdx.x * blockDim.x + threadIdx.x;
    if (e >= E) return;
    int s = src[e], d = dst[e];
    float norm = dinv[s] * ew[e] * dinv[d];
    atomic_add_f32(&out[d], norm * z[s]);
}

// ---------------- launch ----------------

extern "C" void kernel_launch(void* const* d_in, const int* in_sizes, int n_in,
                              void* d_out, int out_size, void* d_ws, size_t ws_size,
                              hipStream_t stream) {
    const float* x  = (const float*)d_in[0];
    const int*   ei = (const int*)d_in[1];
    const float* ea = (const float*)d_in[2];
    const float* W1 = (const float*)d_in[3];
    const float* b1 = (const float*)d_in[4];
    const float* W2 = (const float*)d_in[5];
    const float* b2 = (const float*)d_in[6];
    float* out = (float*)d_out;

    const int N = in_sizes[0] / NF;      // 100000
    const int E = in_sizes[2];           // 3200000
    const int* srcI = ei;                // edge_index[0]
    const int* dstI = ei + E;            // edge_index[1]

    float* ws   = (float*)d_ws;
    float* dinv = ws;                                  // N floats (deg then dinv)
    float* z    = ws + (size_t)N;                      // N floats
    float* xw   = ws + 2 * (size_t)N;                  // N*NH floats
    float* h1   = xw + (size_t)N * NH;                 // N*NH floats

    const int T = 256;

    // normalization coefficients
    k_deg_init <<<(N + T - 1) / T, T, 0, stream>>>(dinv, N);
    k_deg_accum<<<(E + T - 1) / T, T, 0, stream>>>(dstI, ea, dinv, E);
    k_dinv     <<<(N + T - 1) / T, T, 0, stream>>>(dinv, N);

    // layer 1 GEMM (WMMA)
    const int n_tiles = N / 16;
    if (n_tiles > 0)
        k_gemm1_wmma<<<n_tiles, 128, 0, stream>>>(x, W1, xw, n_tiles);
    const int row0 = n_tiles * 16;
    if (row0 < N) {
        int rem = (N - row0) * NH;
        k_gemm1_tail<<<(rem + T - 1) / T, T, 0, stream>>>(x, W1, xw, row0, N);
    }

    // layer 1 aggregation: init (bias + self-loop), then edge scatter, wave/edge
    k_h1_init<<<((size_t)N * NH + T - 1) / T, T, 0, stream>>>(xw, dinv, b1, h1, N);
    {
        long long threads = (long long)E * 32;
        k_scatter1<<<(unsigned)((threads + T - 1) / T), T, 0, stream>>>(
            srcI, dstI, ea, dinv, xw, h1, E);
    }

    // layer 2: relu + dot, init (bias + self-loop), edge scatter
    k_relu_dot<<<(N + T - 1) / T, T, 0, stream>>>(h1, W2, z, N);
    k_out_init<<<(N + T - 1) / T, T, 0, stream>>>(z, dinv, b2, out, N);
    k_scatter2<<<(E + T - 1) / T, T, 0, stream>>>(srcI, dstI, ea, dinv, z, out, E);
}